// StructureFeatureEncoder_27711128994202
// MI455X (gfx1250) — compile-verified
//
#include <hip/hip_runtime.h>
#include <hip/hip_bf16.h>

#define B_ 8
#define N_ 8192
#define E_ 16384
#define D_ 128
#define R_ 200
#define M_ 20
#define L_ 3

typedef __attribute__((ext_vector_type(2))) float v2f;
typedef __attribute__((ext_vector_type(8))) float v8f;

// ---------------------------------------------------------------------------
// h0 = dist_table[clip(dists,0,9)] + 0.1 * noise
// ---------------------------------------------------------------------------
__global__ __launch_bounds__(256)
void init_h_kernel(const int* __restrict__ dists,
                   const float* __restrict__ noise,
                   const float* __restrict__ dist_table,
                   float* __restrict__ h)
{
    size_t i = (size_t)blockIdx.x * blockDim.x + threadIdx.x;
    if (i >= (size_t)B_ * N_ * D_) return;
    int d = (int)(i & (D_ - 1));
    size_t bn = i >> 7;
    int di = dists[bn];
    di = di < 0 ? 0 : (di > 9 ? 9 : di);
    h[i] = dist_table[di * D_ + d] + 0.1f * noise[i];
}

__global__ __launch_bounds__(256)
void zero_kernel(float* __restrict__ p, size_t n)
{
    size_t i = (size_t)blockIdx.x * blockDim.x + threadIdx.x;
    if (i < n) p[i] = 0.0f;
}

// ---------------------------------------------------------------------------
// Message GEMM + relu + edge-mask + fused scatter-add (atomicAdd into aggr).
// One workgroup = 16 edges. 8 waves, wave w owns output columns [16w,16w+16).
// A-tile (16 x 640) built in LDS: [comp | h_src | dist_src | h_r | conf].
// K-loop: 160 x v_wmma_f32_16x16x4_f32.
// ---------------------------------------------------------------------------
__global__ __launch_bounds__(256)
void msg_scatter_kernel(const float* __restrict__ h,
                        const int* __restrict__ edge_index,
                        const int* __restrict__ rels,
                        const int* __restrict__ dists,
                        const unsigned char* __restrict__ edge_mask,
                        const float* __restrict__ conf,
                        const float* __restrict__ rel_table,
                        const float* __restrict__ dist_table,
                        const float* __restrict__ msg_W,
                        const float* __restrict__ msg_b,
                        float* __restrict__ aggr,
                        int layer)
{
    __shared__ float As[16][5 * D_ + 4];
    __shared__ int   s_src[16], s_tgt[16], s_rel[16], s_di[16];
    __shared__ float s_em[16];

    const int t    = threadIdx.x;
    const int tile = blockIdx.x;
    const int b    = tile / (E_ / 16);
    const int e0   = (tile % (E_ / 16)) * 16;

    if (t < 16) {
        int e  = e0 + t;
        int s  = edge_index[(size_t)b * 2 * E_ + e];
        int tg = edge_index[(size_t)b * 2 * E_ + E_ + e];
        s_src[t] = s;
        s_tgt[t] = tg;
        s_rel[t] = rels[(size_t)b * E_ + e];
        int di = dists[(size_t)b * N_ + s];
        s_di[t] = di < 0 ? 0 : (di > 9 ? 9 : di);
        s_em[t] = edge_mask[(size_t)b * E_ + e] ? 1.0f : 0.0f;
    }
    __syncthreads();

    for (int i = t; i < 16 * D_; i += 256) {
        int r = i >> 7;
        int d = i & (D_ - 1);
        float hs = h[((size_t)b * N_ + s_src[r]) * D_ + d];
        float hr = rel_table[(size_t)s_rel[r] * D_ + d];
        float ds = dist_table[(size_t)s_di[r] * D_ + d];
        float cf = conf[((size_t)b * E_ + e0 + r) * D_ + d];
        As[r][0 * D_ + d] = hs * hr;
        As[r][1 * D_ + d] = hs;
        As[r][2 * D_ + d] = ds;
        As[r][3 * D_ + d] = hr;
        As[r][4 * D_ + d] = cf;
    }
    __syncthreads();

    const int wave    = t >> 5;
    const int lane    = t & 31;
    const int m       = lane & 15;
    const int koff    = (lane >> 4) << 1;   // 0 or 2
    const int colbase = wave * 16;
    const float* W = msg_W + (size_t)layer * 5 * D_ * D_;

    v8f acc = {0.f, 0.f, 0.f, 0.f, 0.f, 0.f, 0.f, 0.f};
    for (int kk = 0; kk < 5 * D_; kk += 4) {
        v2f a, bm;
        a.x  = As[m][kk + koff];
        a.y  = As[m][kk + koff + 1];
        bm.x = W[(size_t)(kk + koff) * D_ + colbase + m];
        bm.y = W[(size_t)(kk + koff + 1) * D_ + colbase + m];
        acc = __builtin_amdgcn_wmma_f32_16x16x4_f32(false, a, false, bm,
                                                    (short)0, acc, false, false);
    }

    const int rbase = (lane >> 4) << 3;     // 0 or 8
    const int col   = colbase + m;
    const float bias = msg_b[layer * D_ + col];
#pragma unroll
    for (int v = 0; v < 8; ++v) {
        int row = rbase + v;
        float val = acc[v] + bias;
        val = fmaxf(val, 0.0f) * s_em[row];
        atomicAdd(&aggr[((size_t)b * N_ + s_tgt[row]) * D_ + col], val);
    }
}

// ---------------------------------------------------------------------------
// h = aggr @ upd_W[k] + upd_b[k] + h   (16-row tiles, 8 waves x 16 cols)
// ---------------------------------------------------------------------------
__global__ __launch_bounds__(256)
void update_kernel(float* __restrict__ h,
                   const float* __restrict__ aggr,
                   const float* __restrict__ upd_W,
                   const float* __restrict__ upd_b,
                   int layer)
{
    __shared__ float As[16][D_ + 4];
    const int t = threadIdx.x;
    const size_t r0 = (size_t)blockIdx.x * 16;   // row in flattened B*N

    for (int i = t; i < 16 * D_; i += 256) {
        int r = i >> 7;
        int d = i & (D_ - 1);
        As[r][d] = aggr[(r0 + r) * D_ + d];
    }
    __syncthreads();

    const int wave    = t >> 5;
    const int lane    = t & 31;
    const int m       = lane & 15;
    const int koff    = (lane >> 4) << 1;
    const int colbase = wave * 16;
    const float* W = upd_W + (size_t)layer * D_ * D_;

    v8f acc = {0.f, 0.f, 0.f, 0.f, 0.f, 0.f, 0.f, 0.f};
    for (int kk = 0; kk < D_; kk += 4) {
        v2f a, bm;
        a.x  = As[m][kk + koff];
        a.y  = As[m][kk + koff + 1];
        bm.x = W[(size_t)(kk + koff) * D_ + colbase + m];
        bm.y = W[(size_t)(kk + koff + 1) * D_ + colbase + m];
        acc = __builtin_amdgcn_wmma_f32_16x16x4_f32(false, a, false, bm,
                                                    (short)0, acc, false, false);
    }

    const int rbase = (lane >> 4) << 3;
    const int col   = colbase + m;
    const float bias = upd_b[layer * D_ + col];
#pragma unroll
    for (int v = 0; v < 8; ++v) {
        size_t idx = (r0 + rbase + v) * D_ + col;
        h[idx] = acc[v] + bias + h[idx];
    }
}

// ---------------------------------------------------------------------------
// Attention scores + softmax + top-k(M=20) + gathers. One block per batch.
// ---------------------------------------------------------------------------
__global__ __launch_bounds__(1024)
void finalize_kernel(const float* __restrict__ h,
                     const unsigned char* __restrict__ node_mask,
                     const float* __restrict__ rq,
                     const float* __restrict__ att_W,
                     const float* __restrict__ att_b,
                     float* __restrict__ out)
{
    __shared__ float sc[N_];
    __shared__ float red[1024];
    __shared__ int   redi[1024];
    __shared__ float s_scalar[4];
    __shared__ int   s_selidx;

    const int b = blockIdx.x;
    const int t = threadIdx.x;

    // rq . att_W[D:2D]
    if (t < D_) red[t] = rq[b * D_ + t] * att_W[D_ + t];
    __syncthreads();
    if (t == 0) {
        float s = 0.f;
        for (int i = 0; i < D_; ++i) s += red[i];
        s_scalar[0] = s;
    }
    __syncthreads();
    const float rqdot = s_scalar[0] + att_b[0];

    // scores
    for (int n = t; n < N_; n += 1024) {
        const float* hp = h + ((size_t)b * N_ + n) * D_;
        float dot = 0.f;
        for (int d = 0; d < D_; ++d) dot += hp[d] * att_W[d];
        float s = dot + rqdot;
        s = s >= 0.f ? s : 0.01f * s;             // leaky_relu(0.01)
        sc[n] = node_mask[(size_t)b * N_ + n] ? s : -1.0e9f;
    }
    __syncthreads();

    // max reduction
    float lm = -3.4e38f;
    for (int n = t; n < N_; n += 1024) lm = fmaxf(lm, sc[n]);
    red[t] = lm;
    __syncthreads();
    for (int sred = 512; sred > 0; sred >>= 1) {
        if (t < sred) red[t] = fmaxf(red[t], red[t + sred]);
        __syncthreads();
    }
    if (t == 0) s_scalar[1] = red[0];
    __syncthreads();
    const float maxv = s_scalar[1];

    // sum(exp)
    float ls = 0.f;
    for (int n = t; n < N_; n += 1024) ls += expf(sc[n] - maxv);
    __syncthreads();
    red[t] = ls;
    __syncthreads();
    for (int sred = 512; sred > 0; sred >>= 1) {
        if (t < sred) red[t] += red[t + sred];
        __syncthreads();
    }
    if (t == 0) s_scalar[2] = red[0];
    __syncthreads();
    const float sumexp = s_scalar[2];

    // iterated argmax top-k (softmax is monotone: top-k scores == top-k alpha)
    for (int ksel = 0; ksel < M_; ++ksel) {
        float bv = -3.4e38f;
        int   bi = N_;
        for (int n = t; n < N_; n += 1024) {
            float v = sc[n];
            if (v > bv || (v == bv && n < bi)) { bv = v; bi = n; }
        }
        red[t] = bv;
        redi[t] = bi;
        __syncthreads();
        for (int sred = 512; sred > 0; sred >>= 1) {
            if (t < sred) {
                float ov = red[t + sred];
                int   oi = redi[t + sred];
                if (ov > red[t] || (ov == red[t] && oi < redi[t])) {
                    red[t] = ov;
                    redi[t] = oi;
                }
            }
            __syncthreads();
        }
        if (t == 0) { s_selidx = redi[0]; s_scalar[3] = red[0]; }
        __syncthreads();
        const int   idx   = s_selidx;
        const float alpha = expf(s_scalar[3] - maxv) / sumexp;
        if (t < D_) {
            float hv = h[((size_t)b * N_ + idx) * D_ + t];
            hv = node_mask[(size_t)b * N_ + idx] ? hv : 0.f;
            out[((size_t)b * M_ + ksel) * D_ + t] = hv * alpha;
        }
        if (t == 0) sc[idx] = -3.4e38f;
        __syncthreads();
    }

    // t_state = (h * node_mask)[:, 0, :]
    if (t < D_) {
        float hv = h[(size_t)b * N_ * D_ + t];
        hv = node_mask[(size_t)b * N_] ? hv : 0.f;
        out[(size_t)B_ * M_ * D_ + (size_t)b * D_ + t] = hv;
    }
}

// ---------------------------------------------------------------------------
extern "C" void kernel_launch(void* const* d_in, const int* in_sizes, int n_in,
                              void* d_out, int out_size, void* d_ws, size_t ws_size,
                              hipStream_t stream)
{
    (void)in_sizes; (void)n_in; (void)out_size; (void)ws_size;

    const int*   dists      = (const int*)d_in[0];
    const int*   edge_index = (const int*)d_in[1];
    const int*   rels       = (const int*)d_in[2];
    const unsigned char* node_mask = (const unsigned char*)d_in[3];
    const unsigned char* edge_mask = (const unsigned char*)d_in[4];
    const float* rq         = (const float*)d_in[5];
    const float* conf       = (const float*)d_in[6];
    const float* noise      = (const float*)d_in[7];
    const float* dist_table = (const float*)d_in[8];
    const float* rel_table  = (const float*)d_in[9];
    const float* msg_W      = (const float*)d_in[10];
    const float* msg_b      = (const float*)d_in[11];
    const float* upd_W      = (const float*)d_in[12];
    const float* upd_b      = (const float*)d_in[13];
    const float* att_W      = (const float*)d_in[14];
    const float* att_b      = (const float*)d_in[15];

    float* h    = (float*)d_ws;                    // B*N*D
    float* aggr = h + (size_t)B_ * N_ * D_;        // B*N*D

    const size_t hnd = (size_t)B_ * N_ * D_;
    const int blk256 = (int)((hnd + 255) / 256);

    init_h_kernel<<<blk256, 256, 0, stream>>>(dists, noise, dist_table, h);

    for (int k = 0; k < L_; ++k) {
        zero_kernel<<<blk256, 256, 0, stream>>>(aggr, hnd);
        msg_scatter_kernel<<<(B_ * E_) / 16, 256, 0, stream>>>(
            h, edge_index, rels, dists, edge_mask, conf,
            rel_table, dist_table, msg_W, msg_b, aggr, k);
        update_kernel<<<(B_ * N_) / 16, 256, 0, stream>>>(
            h, aggr, upd_W, upd_b, k);
    }

    finalize_kernel<<<B_, 1024, 0, stream>>>(h, node_mask, rq, att_W, att_b,
                                             (float*)d_out);
}